// UniformBottomUpHTMM_67877663146179
// MI455X (gfx1250) — compile-verified
//
#include <hip/hip_runtime.h>
#include <math.h>

// ---------------------------------------------------------------------------
// UniformBottomUpHTMM on MI455X (gfx1250, wave32).
//   C=16 states, M=64 symbols, G=16 groups, T=64 trees, depth D=10.
//   Per level: per-group 16x16 transition x 16-edge state block via
//   V_WMMA_F32_16X16X4_F32 chained 4x (K=16). Heap layout => sibling pair-sum
//   happens inside the WMMA accumulator registers.
//   Operands are single b64 loads per lane (half-wave offset folded into the
//   address). Half-wave reductions use DS_SWIZZLE_B32 xor mode (immediate
//   lane pattern -> 1 ds + 1 add per step, no mbcnt/cndmask address math).
// ---------------------------------------------------------------------------

namespace {
constexpr int C = 16;
constexpr int M = 64;
constexpr int G = 16;
constexpr int T = 64;
constexpr int D = 10;
constexpr int N1 = (1 << (D + 1)) - 1;      // 2047 nodes per tree
constexpr long DIM = (long)T * N1;          // 131008 nodes total
constexpr int NLEAVES = T << D;             // 65536

typedef __attribute__((ext_vector_type(2))) float v2f;
typedef __attribute__((ext_vector_type(8))) float v8f;
} // namespace

// ds_swizzle_b32, group-of-32 mode: lane' = ((lane & 0x1f) | 0) ^ XOR
// Sum across 16 lanes (xor masks 1,2,4,8 never cross the half-wave boundary).
template <int XOR>
__device__ __forceinline__ float swz_xor(float v) {
  return __int_as_float(
      __builtin_amdgcn_ds_swizzle(__float_as_int(v), (XOR << 10) | 0x1f));
}
__device__ __forceinline__ float half_wave_sum(float v) {
  v += swz_xor<1>(v);
  v += swz_xor<2>(v);
  v += swz_xor<4>(v);
  v += swz_xor<8>(v);
  return v;
}

// ---------------------------------------------------------------------------
// Softmax prep.
//   ATW: WMMA-ready B-operand pack of A^T. For chained-k wmma and lane half h,
//        lane n loads {A^T[4k+2h][n], A^T[4k+2h+1][n]} as one float2:
//        ATW[((g*4 + k)*2 + h)*16 + n][0..1]
//   Bms[g][m][c] = softmax_m(lambda_B[c,m,g])      (lane c contiguous)
//   Pis[g][c]    = softmax_c(lambda_Pi[c,g])
// ---------------------------------------------------------------------------
__global__ void prep_softmax(const float* __restrict__ lamA,
                             const float* __restrict__ lamB,
                             const float* __restrict__ lamPi,
                             float* __restrict__ ATW,
                             float* __restrict__ Bms,
                             float* __restrict__ Pis) {
  int tid = blockIdx.x * blockDim.x + threadIdx.x;
  if (tid < C * G) {                       // transition softmax over i (axis 0)
    int j = tid >> 4, g = tid & 15;        // this thread owns column j of A_g
    float v[C];
    float mx = -3.4e38f;
    for (int i = 0; i < C; ++i) {
      v[i] = lamA[((long)i * C + j) * G + g];
      mx = fmaxf(mx, v[i]);
    }
    float s = 0.f;
    for (int i = 0; i < C; ++i) { v[i] = expf(v[i] - mx); s += v[i]; }
    float inv = 1.f / s;
    // j = 4k + 2*half + lo  ->  scatter into the wmma-ready pack
    int k = j >> 2, half = (j >> 1) & 1, lo = j & 1;
    for (int i = 0; i < C; ++i)
      ATW[((((long)g * 4 + k) * 2 + half) * 16 + i) * 2 + lo] = v[i] * inv;
  } else if (tid < 2 * C * G) {            // emission softmax over m (axis 1)
    int p = tid - C * G;
    int c = p >> 4, g = p & 15;
    float mx = -3.4e38f;
    for (int m = 0; m < M; ++m) mx = fmaxf(mx, lamB[((long)c * M + m) * G + g]);
    float s = 0.f;
    for (int m = 0; m < M; ++m) s += expf(lamB[((long)c * M + m) * G + g] - mx);
    float inv = 1.f / s;
    for (int m = 0; m < M; ++m)
      Bms[((long)g * M + m) * C + c] = expf(lamB[((long)c * M + m) * G + g] - mx) * inv;
  } else if (tid < 2 * C * G + G) {        // prior softmax over c (axis 0)
    int g = tid - 2 * C * G;
    float v[C];
    float mx = -3.4e38f;
    for (int c = 0; c < C; ++c) { v[c] = lamPi[(long)c * G + g]; mx = fmaxf(mx, v[c]); }
    float s = 0.f;
    for (int c = 0; c < C; ++c) { v[c] = expf(v[c] - mx); s += v[c]; }
    float inv = 1.f / s;
    for (int c = 0; c < C; ++c) Pis[(long)g * C + c] = v[c] * inv;
  }
}

// ---------------------------------------------------------------------------
// Leaves: beta[g][leaf][c] = Pi[g][c]*Bm[g][x][c] / nu ;  ll[g][leaf] = log(nu)
// One half-wave (16 lanes = 16 states) per (leaf, g). Coalesced 16-float rows.
// ---------------------------------------------------------------------------
__global__ void leaf_kernel(const int* __restrict__ x,
                            const int* __restrict__ inv_map,
                            const int* __restrict__ leaves,
                            const float* __restrict__ Bms,
                            const float* __restrict__ Pis,
                            float* __restrict__ beta,
                            float* __restrict__ llw) {
  int tid = blockIdx.x * blockDim.x + threadIdx.x;
  int c = tid & 15;
  int pair = tid >> 4;              // = leaf_index * G + g
  int g = pair & 15;
  int li = pair >> 4;
  if (li >= NLEAVES) return;
  int node = leaves[li];
  int xv = x[inv_map[node]];
  float v = Pis[(long)g * C + c] * Bms[((long)g * M + xv) * C + c];
  float nu = half_wave_sum(v);
  beta[((size_t)g * DIM + node) * C + c] = v / nu;
  if (c == 0) llw[(size_t)g * DIM + node] = logf(nu);
}

// ---------------------------------------------------------------------------
// One upward level. One wave = one (16-edge block, group g).
//   A-operand (16 edges x K=16): lane L holds child row (L&15); per wmma k,
//     one b64 load at byte offset 16k + 8*half  (half folded into address).
//   B-operand: one b64 load per k from the ATW pack.
//   4x V_WMMA_F32_16X16X4_F32 accumulate D[edge][i] = sum_j beta[ch,j]*A[i,j].
//   Sibling edges are adjacent D rows -> parent value = (d[2q]+d[2q+1])/2.
// ---------------------------------------------------------------------------
__global__ void level_kernel(const int* __restrict__ x,
                             const int* __restrict__ inv_map,
                             const int* __restrict__ children,
                             const int* __restrict__ uparents,
                             const float* __restrict__ ATW,
                             const float* __restrict__ Bms,
                             float* __restrict__ beta,
                             float* __restrict__ llw,
                             int e0, int u0, int nEdges) {
  int unit = (int)((blockIdx.x * blockDim.x + threadIdx.x) >> 5); // wave id
  int lane = threadIdx.x & 31;
  if (unit >= nEdges) return;               // nUnits == (nEdges/16)*G == nEdges
  int g = unit & 15;
  int eb = (unit >> 4) << 4;                // edge base within level

  int half = lane >> 4;                     // 0: lanes 0-15, 1: lanes 16-31
  int hl = lane & 15;

  // address bases with half-wave K-offset folded in (no register selects)
  int ch = children[e0 + eb + hl];
  const float* arow = beta + ((size_t)g * DIM + ch) * C + 2 * half;
  const float* brow = ATW + (((size_t)g * 8 + half) * 16 + hl) * 2;

  v8f acc = {};
  #pragma unroll
  for (int k = 0; k < 4; ++k) {             // chain K=16 as 4 x (K=4)
    v2f a = *(const v2f*)(arow + 4 * k);    // {beta[ch, 4k+2h], beta[ch, 4k+2h+1]}
    v2f b = *(const v2f*)(brow + 64 * k);   // {A^T[4k+2h][hl], A^T[4k+2h+1][hl]}
    acc = __builtin_amdgcn_wmma_f32_16x16x4_f32(
        /*neg_a=*/false, a, /*neg_b=*/false, b,
        /*c_mod=*/(short)0, acc, /*reuse_a=*/false, /*reuse_b=*/false);
  }

  // --- sibling pair-average + emission + normalize + log, 8 parents/wave ---
  #pragma unroll
  for (int q = 0; q < 4; ++q) {
    float tup = (acc[2 * q] + acc[2 * q + 1]) * 0.5f;   // cnt == 2 everywhere
    int pidx = (eb >> 1) + q + (half ? 4 : 0);
    int upn = uparents[u0 + pidx];
    int xv = x[inv_map[upn]];
    float val = tup * Bms[((long)g * M + xv) * C + hl];
    float nu = half_wave_sum(val);
    beta[((size_t)g * DIM + upn) * C + hl] = val / nu;
    if (hl == 0) llw[(size_t)g * DIM + upn] = logf(nu);
  }
}

// ---------------------------------------------------------------------------
// out[t*G+g] = sum over the 2047 nodes of tree t of ll[g][node].
// One 256-thread block per (t,g); deterministic fixed-order tree reduction.
// ---------------------------------------------------------------------------
__global__ void reduce_kernel(const float* __restrict__ llw,
                              float* __restrict__ out) {
  int t = blockIdx.x >> 4;
  int g = blockIdx.x & 15;
  const float* p = llw + (size_t)g * DIM + (size_t)t * N1;
  float s = 0.f;
  for (int i = threadIdx.x; i < N1; i += blockDim.x) s += p[i];
  __shared__ float sm[256];
  sm[threadIdx.x] = s;
  __syncthreads();
  for (int w = 128; w > 0; w >>= 1) {
    if (threadIdx.x < w) sm[threadIdx.x] += sm[threadIdx.x + w];
    __syncthreads();
  }
  if (threadIdx.x == 0) out[blockIdx.x] = sm[0];
}

// ---------------------------------------------------------------------------
extern "C" void kernel_launch(void* const* d_in, const int* in_sizes, int n_in,
                              void* d_out, int out_size, void* d_ws, size_t ws_size,
                              hipStream_t stream) {
  (void)in_sizes; (void)n_in; (void)out_size; (void)ws_size;
  const int* x            = (const int*)d_in[0];
  const int* leaves       = (const int*)d_in[1];
  const int* trees_ind    = (const int*)d_in[2];   (void)trees_ind;   // implied by heap layout
  const int* parents_flat = (const int*)d_in[3];   (void)parents_flat;// siblings adjacent => implied
  const int* children     = (const int*)d_in[4];
  const int* uparents     = (const int*)d_in[5];
  const int* inv_map      = (const int*)d_in[6];
  const float* lamA       = (const float*)d_in[7];
  const float* lamB       = (const float*)d_in[8];
  const float* lamPi      = (const float*)d_in[9];
  float* out = (float*)d_out;

  // workspace partition (floats). beta = 134 MB (fits MI455X 192 MB L2).
  float* ws   = (float*)d_ws;
  float* ATW  = ws;                                  // G*C*C      = 4096
  float* Bms  = ws + 4096;                           // G*M*C      = 16384
  float* Pis  = ws + 20480;                          // G*C        = 256
  float* beta = ws + 32768;                          // G*DIM*C    = 33538048
  float* llw  = beta + (size_t)G * DIM * C;          // G*DIM      = 2096128

  prep_softmax<<<3, 256, 0, stream>>>(lamA, lamB, lamPi, ATW, Bms, Pis);

  {
    long threads = (long)NLEAVES * G * 16;           // half-wave per (leaf,g)
    leaf_kernel<<<(unsigned)(threads / 256), 256, 0, stream>>>(
        x, inv_map, leaves, Bms, Pis, beta, llw);
  }

  long eoff[D + 1], uoff[D + 1];
  eoff[0] = uoff[0] = 0;
  for (int l = 1; l <= D; ++l) {
    eoff[l] = eoff[l - 1] + ((long)T << l);
    uoff[l] = uoff[l - 1] + ((long)T << (l - 1));
  }

  for (int l = D; l >= 1; --l) {
    int nE = T << l;                                 // edges this level
    long threads = (long)nE * 32;                    // one wave per (16-edge blk, g)
    level_kernel<<<(unsigned)(threads / 256), 256, 0, stream>>>(
        x, inv_map, children, uparents, ATW, Bms, beta, llw,
        (int)eoff[l - 1], (int)uoff[l - 1], nE);
  }

  reduce_kernel<<<T * G, 256, 0, stream>>>(llw, out);
}